// WKPooling_18442589569078
// MI455X (gfx1250) — compile-verified
//
#include <hip/hip_runtime.h>
#include <hip/hip_bf16.h>

// ---- problem constants (from reference setup_inputs) ----
#define NL_          13
#define LAYER_START_ 4
#define LAY_         9      // 13 - 4
#define B_           8
#define T_           512
#define H_           768
#define NTOK_        511    // T - 1
#define WIN_         2

#define LDS_STRIDE 772      // 768 + 4 floats padding -> kills 64-bank conflicts across rows

typedef __attribute__((ext_vector_type(2))) float v2f;
typedef __attribute__((ext_vector_type(8))) float v8f;

// ---------------------------------------------------------------------------
// Kernel 1: one workgroup per token. Computes 9x9 Gram via V_WMMA_F32_16X16X4_F32,
// then per-layer-window Cholesky (== QR's R up to row signs, to which the
// outputs are invariant), producing alpha[9] and var_tok per token.
// ---------------------------------------------------------------------------
__global__ __launch_bounds__(256) void wk_token_kernel(
    const float* __restrict__ hs,     // [NL, B, T, H]
    float* __restrict__ alpha_ws,     // [B*NTOK, LAY]
    float* __restrict__ var_ws)       // [B*NTOK]
{
  __shared__ float sf[16 * LDS_STRIDE];   // 16 rows (rows 9..15 zero), padded
  __shared__ float part[8 * 256];         // 8 partial 16x16 tiles
  __shared__ float G[16 * 16];            // final Gram (only [0..8][0..8] used)
  __shared__ float alignv[LAY_], novv[LAY_], alphav[LAY_];

  const int tok  = blockIdx.x;            // 0 .. B*NTOK-1
  const int b    = tok / NTOK_;
  const int t    = tok % NTOK_;
  const int tid  = threadIdx.x;
  const int lane = tid & 31;
  const int wave = tid >> 5;

  // ---- stage S (9 x 768) into LDS with float4 loads; zero-pad rows 9..15 ----
  for (int idx = tid; idx < 16 * (H_ / 4); idx += 256) {
    const int k  = idx / (H_ / 4);
    const int c4 = idx % (H_ / 4);
    float4 v;
    if (k < LAY_) {
      const float4* src = reinterpret_cast<const float4*>(
          hs + (((size_t)(k + LAYER_START_) * B_ + b) * (size_t)T_ + t) * (size_t)H_);
      v = src[c4];
    } else {
      v = make_float4(0.f, 0.f, 0.f, 0.f);
    }
    reinterpret_cast<float4*>(&sf[k * LDS_STRIDE])[c4] = v;
  }
  __syncthreads();

  // ---- Gram via WMMA: G = S * S^T, padded to 16x16, K stepped by 4 ----
  // A fragment layout (f32 16x16x4): lanes 0-15 hold row M=lane with K=0,1 in
  // v0,v1; lanes 16-31 hold row M=lane-16 with K=2,3. For B = A^T the B-layout
  // (lane = column, K split across lane halves) holds identical register
  // contents, so we pass the same fragment for A and B.
  const int row = lane & 15;
  const int kk  = (lane >> 4) << 1;       // 0 for lanes 0-15, 2 for lanes 16-31
  const float* rp = &sf[row * LDS_STRIDE];

  v8f c = {0.f, 0.f, 0.f, 0.f, 0.f, 0.f, 0.f, 0.f};
  for (int it = 0; it < 24; ++it) {       // each of 8 waves covers 96 columns
    const int h0 = wave * 96 + it * 4;
    v2f a;
    a.x = rp[h0 + kk];
    a.y = rp[h0 + kk + 1];
    c = __builtin_amdgcn_wmma_f32_16x16x4_f32(
        /*neg_a=*/false, a, /*neg_b=*/false, a,
        /*c_mod=*/(short)0, c, /*reuse_a=*/false, /*reuse_b=*/false);
  }
#pragma unroll
  for (int j = 0; j < 8; ++j) part[wave * 256 + j * 32 + lane] = c[j];
  __syncthreads();

  // reduce the 8 partial tiles and scatter into G[M][N]
  {
    float g = 0.f;
#pragma unroll
    for (int w = 0; w < 8; ++w) g += part[w * 256 + tid];
    const int j = tid >> 5;
    const int l = tid & 31;
    const int M = j + ((l >= 16) ? 8 : 0);  // C layout: VGPR j -> M=j / M=j+8
    const int N = l & 15;
    G[M * 16 + N] = g;
  }
  __syncthreads();

  // ---- per-k window math (threads 0..8), m <= 5 ----
  if (tid < LAY_) {
    const int k = tid;
    int rows[5];
    int m = 0;
    if (k >= WIN_) { rows[m++] = k - 2; rows[m++] = k - 1; }     // left (empty if k<2)
    for (int rr = k + 1; rr < k + WIN_ + 1 && rr < LAY_; ++rr)   // right
      rows[m++] = rr;
    rows[m++] = k;                                               // self last

    float Gk[5][5];
    for (int i = 0; i < m; ++i)
      for (int j = 0; j < m; ++j)
        Gk[i][j] = G[rows[i] * 16 + rows[j]];

    // Cholesky: upper-triangular R with R^T R = Gk  (same as QR's R up to row signs)
    float R[5][5];
    for (int i = 0; i < 5; ++i)
      for (int j = 0; j < 5; ++j) R[i][j] = 0.f;
    for (int j = 0; j < m; ++j) {
      for (int i = 0; i < j; ++i) {
        float s = Gk[i][j];
        for (int p = 0; p < i; ++p) s -= R[p][i] * R[p][j];
        R[i][j] = s / R[i][i];
      }
      float s = Gk[j][j];
      for (int p = 0; p < j; ++p) s -= R[p][j] * R[p][j];
      R[j][j] = sqrtf(fmaxf(s, 1e-30f));
    }

    // r = last column of R
    float r[5];
    float rn2 = 0.f, rn2m1 = 0.f;
    for (int i = 0; i < m; ++i) { r[i] = R[i][m - 1]; rn2 += r[i] * r[i]; }
    for (int i = 0; i < m - 1; ++i) rn2m1 += r[i] * r[i];

    // column-normalize Rsub, take row means, dot with r[:-1]
    float coln[4];
    for (int j = 0; j < m - 1; ++j) {
      float cn = 0.f;
      for (int i = 0; i < m - 1; ++i) cn += R[i][j] * R[i][j];
      coln[j] = fmaxf(sqrtf(cn), 1e-12f);
    }
    float acc = 0.f;
    for (int i = 0; i < m - 1; ++i) {
      float rm = 0.f;
      for (int j = 0; j < m - 1; ++j) rm += R[i][j] / coln[j];
      rm /= (float)(m - 1);
      acc += rm * r[i];
    }
    float align = acc / sqrtf(rn2m1);
    align = 1.f / (align * (float)m * 2.f);
    alignv[k] = align;
    novv[k]   = fabsf(r[m - 1]) / sqrtf(rn2);
  }
  __syncthreads();

  // ---- normalize alpha; cosine-sim variance from Gram diagonals/subdiagonal ----
  if (tid == 0) {
    float sa = 0.f, sn = 0.f;
    for (int k = 0; k < LAY_; ++k) { sa += alignv[k]; sn += novv[k]; }
    float s2 = 0.f;
    for (int k = 0; k < LAY_; ++k) {
      const float al = alignv[k] / sa + novv[k] / sn;
      alphav[k] = al;
      s2 += al;
    }
    for (int k = 0; k < LAY_; ++k) alphav[k] /= s2;

    float sims[LAY_ - 1];
    float mean = 0.f;
    for (int l = 0; l < LAY_ - 1; ++l) {
      const float num = G[(l + 1) * 16 + l];
      const float den = fmaxf(sqrtf(G[l * 16 + l]) * sqrtf(G[(l + 1) * 16 + (l + 1)]), 1e-8f);
      sims[l] = num / den;
      mean += sims[l];
    }
    mean /= (float)(LAY_ - 1);
    float v = 0.f;
    for (int l = 0; l < LAY_ - 1; ++l) { const float d = sims[l] - mean; v += d * d; }
    var_ws[tok] = v / (float)(LAY_ - 2);   // ddof=1 over 8 samples
  }
  __syncthreads();

  if (tid < LAY_) alpha_ws[(size_t)tok * LAY_ + tid] = alphav[tid];
}

// ---------------------------------------------------------------------------
// Kernel 2: per-batch sum of var_tok (normalization folded into final divide).
// ---------------------------------------------------------------------------
__global__ __launch_bounds__(256) void wk_varsum_kernel(
    const float* __restrict__ var_ws, float* __restrict__ var_sum)
{
  __shared__ float red[256];
  const int b = blockIdx.x;
  float s = 0.f;
  for (int t = threadIdx.x; t < NTOK_; t += 256) s += var_ws[b * NTOK_ + t];
  red[threadIdx.x] = s;
  __syncthreads();
  for (int off = 128; off > 0; off >>= 1) {
    if (threadIdx.x < off) red[threadIdx.x] += red[threadIdx.x + off];
    __syncthreads();
  }
  if (threadIdx.x == 0) var_sum[b] = red[0];
}

// ---------------------------------------------------------------------------
// Kernel 3: out[b,h] = sum_t var[t] * sum_k alpha[t,k]*sf[b,t,k,h] / var_sum[b].
// Input re-read is L2-resident (163 MB fits in 192 MB L2). Coalesced over h.
// ---------------------------------------------------------------------------
__global__ __launch_bounds__(256) void wk_out_kernel(
    const float* __restrict__ hs, const float* __restrict__ alpha_ws,
    const float* __restrict__ var_ws, const float* __restrict__ var_sum,
    float* __restrict__ out)
{
  const int gid = blockIdx.x * 256 + threadIdx.x;   // 0 .. B*H-1
  const int b = gid / H_;
  const int h = gid % H_;
  float acc = 0.f;
  for (int t = 0; t < NTOK_; ++t) {
    const int tok = b * NTOK_ + t;
    const float vw = var_ws[tok];
    float e = 0.f;
#pragma unroll
    for (int k = 0; k < LAY_; ++k) {
      const float a = alpha_ws[(size_t)tok * LAY_ + k];        // uniform broadcast
      const float s = hs[(((size_t)(k + LAYER_START_) * B_ + b) * T_ + t) * (size_t)H_ + h];
      e += a * s;
    }
    acc += vw * e;
  }
  out[gid] = acc / var_sum[b];
}

// ---------------------------------------------------------------------------
extern "C" void kernel_launch(void* const* d_in, const int* in_sizes, int n_in,
                              void* d_out, int out_size, void* d_ws, size_t ws_size,
                              hipStream_t stream) {
  const float* hs = (const float*)d_in[0];   // all_hidden_states, fp32
  // d_in[1] (attention_mask) is unused by the reference.

  float* alpha_ws = (float*)d_ws;                              // B*NTOK*LAY floats
  float* var_ws   = alpha_ws + (size_t)B_ * NTOK_ * LAY_;      // B*NTOK floats
  float* var_sum  = var_ws + (size_t)B_ * NTOK_;               // B floats

  wk_token_kernel<<<B_ * NTOK_, 256, 0, stream>>>(hs, alpha_ws, var_ws);
  wk_varsum_kernel<<<B_, 256, 0, stream>>>(var_ws, var_sum);
  wk_out_kernel<<<(B_ * H_) / 256, 256, 0, stream>>>(hs, alpha_ws, var_ws, var_sum,
                                                     (float*)d_out);
}